// TextureMartingaleModule_2199023256199
// MI455X (gfx1250) — compile-verified
//
#include <hip/hip_runtime.h>

// Texture-martingale 3x3 stencil: x(4,16,512,512) f32 -> out(4,64,512,512) f32.
// HBM floor ~335MB -> ~14us @23.3TB/s. Halo tiles staged via TDM
// tensor_load_to_lds (one row-iterated descriptor per block, TENSORcnt path);
// logs hoisted to a shared LDS tile; sliding row aggregates for box sums.

#define HN 512
#define TILE_W 64
#define TILE_H 16
#define HALO_W (TILE_W + 2)          // 66
#define HALO_H (TILE_H + 2)          // 18
#define LDS_N  (HALO_W * HALO_H)     // 1188 floats

typedef __attribute__((address_space(3))) float lds_f32;
typedef unsigned int u32x4 __attribute__((ext_vector_type(4)));
typedef int          i32x4 __attribute__((ext_vector_type(4)));
typedef int          i32x8 __attribute__((ext_vector_type(8)));

__global__ __launch_bounds__(256)
void texture_martingale_kernel(const float* __restrict__ in, float* __restrict__ out)
{
    __shared__ float traw[LDS_N];    // raw halo tile (zero padded)
    __shared__ float tlog[LDS_N];    // v * log(v + eps), shared across 9 windows

    const int tid = threadIdx.x;
    const int p   = blockIdx.z;                 // plane = b*16 + c  (0..63)
    const int h0  = blockIdx.y * TILE_H;
    const int w0  = blockIdx.x * TILE_W;
    const float* src = in + (size_t)p * (HN * HN);

    const float EPS = 1e-6f;

    // ---- Phase 1: zero-fill raw tile (provides zero padding at edges) ----
    for (int i = tid; i < LDS_N; i += 256) traw[i] = 0.0f;
    __syncthreads();

    // ---- Phase 2: TDM tensor load of the clipped halo region (wave 0) ----
    if (tid < 32) {
        const int xs = (w0 > 0) ? (w0 - 1) : 0;
        const int xe = (w0 + TILE_W < HN) ? (w0 + TILE_W) : (HN - 1);
        const int ys = (h0 > 0) ? (h0 - 1) : 0;
        const int ye = (h0 + TILE_H < HN) ? (h0 + TILE_H) : (HN - 1);
        const int width = xe - xs + 1;          // 64..66 valid columns
        const int nrows = ye - ys + 1;          // 16..18 valid rows
        const int lds_elem = (ys - (h0 - 1)) * HALO_W + (xs - (w0 - 1));
        const unsigned lds_byte =
            (unsigned)(unsigned long long)(lds_f32*)&traw[0] + 4u * (unsigned)lds_elem;
        const unsigned long long ga =
            (unsigned long long)(src + (size_t)ys * HN + xs);

        // D# group0: count=1 | lds_addr | global_addr(57b) | type=2
        u32x4 g0 = { 1u, lds_byte, (unsigned)ga,
                     (unsigned)(ga >> 32) | 0x80000000u };
        // D# group1: data_size=4B(code 2), iterate_enable=1;
        // tensor_dim0 = tensor_dim1 = 1M (OOB logic never consulted -> explicit clip);
        // tile_dim0 = width, tile_dim1 = 1 row per iteration, stride0 = 512.
        i32x8 g1 = { (2 << 16) | (1 << 19),
                     0,                         // abar=0, tensor_dim0[15:0]=0
                     0x0010,                    // tensor_dim0[31:16]=16 (1M); dim1 lo=0
                     0x0010 | (width << 16),    // tensor_dim1[31:16]=16 (1M); tile_dim0
                     1,                         // tile_dim1=1, tile_dim2=0
                     HN,                        // tensor_dim0_stride = 512
                     0, 0 };
        // D# group2 (iterate mode): lds_inc=66 elems, global_inc=512 elems,
        // iterate_count = nrows-1 ("0 = 1 iteration" encoding)
        i32x4 g2 = { 0, HALO_W, HN, (nrows - 1) << 16 };
        i32x4 g3 = { 0, 0, 0, 0 };              // group3 unused (zeros)
        i32x8 g4 = { 0, 0, 0, 0, 0, 0, 0, 0 };  // extra arg of 6-arg toolchain form

        __builtin_amdgcn_tensor_load_to_lds(g0, g1, g2, g3, g4, 0);
        __builtin_amdgcn_s_wait_tensorcnt(0);
    }
    __syncthreads();

    // ---- Phase 2b: pointwise log transform, once per tile element ----
    for (int i = tid; i < LDS_N; i += 256) {
        float v = traw[i];
        tlog[i] = v * __logf(v + EPS);          // v_log_f32 path; v==0 -> 0
    }
    __syncthreads();

    // ---- Phase 3: 4 rows per thread, sliding raw window + row aggregates ----
    const int tx  = tid & (TILE_W - 1);
    const int ty0 = (tid >> 6) * 4;             // rows ty0..ty0+3

    const float INV9      = 1.0f / 9.0f;
    const float INV8      = 1.0f / 8.0f;
    const float SCALE     = 0.60653065971263342f;   // exp(-0.5), theta == 1
    const float EPS_SCALE = EPS * SCALE;

#define TRAW(r, c) traw[(r) * HALO_W + (c)]
#define TLOG(r, c) tlog[(r) * HALO_W + (c)]

    float w00, w01, w02, w10, w11, w12, w20, w21, w22;   // raw 3x3 window
    float rs0, rs1, rs2;   // per-row 3-tap sum of v
    float rq0, rq1, rq2;   // per-row 3-tap sum of v^2
    float rl0, rl1, rl2;   // per-row 3-tap sum of v*log(v+eps)

    w00 = TRAW(ty0 + 0, tx + 0); w01 = TRAW(ty0 + 0, tx + 1); w02 = TRAW(ty0 + 0, tx + 2);
    w10 = TRAW(ty0 + 1, tx + 0); w11 = TRAW(ty0 + 1, tx + 1); w12 = TRAW(ty0 + 1, tx + 2);
    rs0 = w00 + w01 + w02;
    rq0 = __builtin_fmaf(w02, w02, __builtin_fmaf(w01, w01, w00 * w00));
    rl0 = TLOG(ty0 + 0, tx + 0) + TLOG(ty0 + 0, tx + 1) + TLOG(ty0 + 0, tx + 2);
    rs1 = w10 + w11 + w12;
    rq1 = __builtin_fmaf(w12, w12, __builtin_fmaf(w11, w11, w10 * w10));
    rl1 = TLOG(ty0 + 1, tx + 0) + TLOG(ty0 + 1, tx + 1) + TLOG(ty0 + 1, tx + 2);

    const size_t plane_stride = (size_t)HN * HN;

#pragma unroll
    for (int r = 0; r < 4; ++r) {
        const int ty = ty0 + r;
        w20 = TRAW(ty + 2, tx + 0); w21 = TRAW(ty + 2, tx + 1); w22 = TRAW(ty + 2, tx + 2);
        rs2 = w20 + w21 + w22;
        rq2 = __builtin_fmaf(w22, w22, __builtin_fmaf(w21, w21, w20 * w20));
        rl2 = TLOG(ty + 2, tx + 0) + TLOG(ty + 2, tx + 1) + TLOG(ty + 2, tx + 2);

        const float s  = rs0 + rs1 + rs2;
        const float s2 = rq0 + rq1 + rq2;
        const float pl = rl0 + rl1 + rl2;
        const float mean = s * INV9;

        // centered terms (exact two-pass like the reference; c regs shared)
        const float c0 = w00 - mean, c1 = w01 - mean, c2 = w02 - mean;
        const float c3 = w10 - mean, c4 = w11 - mean, c5 = w12 - mean;
        const float c6 = w20 - mean, c7 = w21 - mean, c8 = w22 - mean;
        float css = c0 * c0;
        css = __builtin_fmaf(c1, c1, css); css = __builtin_fmaf(c2, c2, css);
        css = __builtin_fmaf(c3, c3, css); css = __builtin_fmaf(c4, c4, css);
        css = __builtin_fmaf(c5, c5, css); css = __builtin_fmaf(c6, c6, css);
        css = __builtin_fmaf(c7, c7, css); css = __builtin_fmaf(c8, c8, css);
        const float sad = ((__builtin_fabsf(c0) + __builtin_fabsf(c1)) +
                           (__builtin_fabsf(c2) + __builtin_fabsf(c3))) +
                          ((__builtin_fabsf(c4) + __builtin_fabsf(c5)) +
                           (__builtin_fabsf(c6) + __builtin_fabsf(c7))) +
                          __builtin_fabsf(c8);

        const float stdv     = __builtin_amdgcn_sqrtf(css * INV8) + EPS;
        const float contrast = (css * INV9) * __builtin_amdgcn_rcpf(stdv * stdv);
        const float energy   = s2 * INV9;
        const float entropy  = -pl * INV9;
        const float homog    = __builtin_amdgcn_rcpf(__builtin_fmaf(sad, INV9, 1.0f));

        const int gy = h0 + ty;
        const int gx = w0 + tx;
        const size_t o = ((size_t)(p * 4) * HN + gy) * HN + gx;  // (p*4+f) planes
        out[o + 0 * plane_stride] = __builtin_fmaf(contrast, SCALE, EPS_SCALE);
        out[o + 1 * plane_stride] = __builtin_fmaf(energy,   SCALE, EPS_SCALE);
        out[o + 2 * plane_stride] = __builtin_fmaf(entropy,  SCALE, EPS_SCALE);
        out[o + 3 * plane_stride] = __builtin_fmaf(homog,    SCALE, EPS_SCALE);

        // slide window + aggregates down one row
        w00 = w10; w01 = w11; w02 = w12;
        w10 = w20; w11 = w21; w12 = w22;
        rs0 = rs1; rs1 = rs2;
        rq0 = rq1; rq1 = rq2;
        rl0 = rl1; rl1 = rl2;
    }
#undef TRAW
#undef TLOG
}

extern "C" void kernel_launch(void* const* d_in, const int* in_sizes, int n_in,
                              void* d_out, int out_size, void* d_ws, size_t ws_size,
                              hipStream_t stream) {
    (void)in_sizes; (void)n_in; (void)out_size; (void)d_ws; (void)ws_size;
    const float* x = (const float*)d_in[0];
    float* out = (float*)d_out;
    dim3 grid(HN / TILE_W, HN / TILE_H, 64);   // 8 x 32 x 64 blocks
    dim3 block(256);                           // 8 wave32s
    texture_martingale_kernel<<<grid, block, 0, stream>>>(x, out);
}